// GNS_67594195304972
// MI455X (gfx1250) — compile-verified
//
#include <hip/hip_runtime.h>

// CDNA5 / gfx1250, wave32.
typedef __attribute__((ext_vector_type(2))) float v2f;
typedef __attribute__((ext_vector_type(8))) float v8f;

// losses = sum_k 2*(K-k) * [ sum_n W3_k * leaky(W2_k * leaky(W1_k . x_n + b1_k) + b2_k) + N*b3_k ]
// Layer 1 (the only real GEMM) runs on V_WMMA_F32_16X16X4_F32 with the bias
// folded into the C accumulator; layers 2/3 are diagonal -> per-lane VALU.

__global__ void __launch_bounds__(256) mlp_wmma_kernel(
    const float* __restrict__ input,   // N*2 (row-major float2)
    const float* __restrict__ W1,      // K*2
    const float* __restrict__ B1,      // K
    const float* __restrict__ W2,      // K
    const float* __restrict__ B2,      // K
    const float* __restrict__ W3,      // K
    float* __restrict__ partials,      // one per workgroup
    int N, int K)
{
    const int lane = threadIdx.x & 31;
    const int wave = threadIdx.x >> 5;
    const int j    = lane & 15;
    const bool lo  = lane < 16;

    // D-matrix layout: lane L owns column (block) L&15; WMMA0 -> blocks 0..15,
    // WMMA1 -> blocks 16..31 (columns >= K zero-padded, c_k = 0).
    const int  jb0 = j;
    const int  jb1 = 16 + j;
    const bool v1  = jb1 < K;

    // B matrix 4x16 f32 (2 VGPRs): VGPR0 = row0 (lanes 0-15) | row2=0 (lanes 16-31)
    //                              VGPR1 = row1 (lanes 0-15) | row3=0 (lanes 16-31)
    v2f bm0, bm1;
    bm0.x = lo ? W1[2 * jb0 + 0] : 0.0f;
    bm0.y = lo ? W1[2 * jb0 + 1] : 0.0f;
    bm1.x = (lo && v1) ? W1[2 * jb1 + 0] : 0.0f;
    bm1.y = (lo && v1) ? W1[2 * jb1 + 1] : 0.0f;

    // Per-lane per-block constants for the diagonal layers.
    const float b1c0 = B1[jb0];
    const float w2c0 = W2[jb0];
    const float b2c0 = B2[jb0];
    const float cc0  = 2.0f * (float)(K - jb0) * W3[jb0];
    const float b1c1 = v1 ? B1[jb1] : 0.0f;
    const float w2c1 = v1 ? W2[jb1] : 0.0f;
    const float b2c1 = v1 ? B2[jb1] : 0.0f;
    const float cc1  = v1 ? (2.0f * (float)(K - jb1) * W3[jb1]) : 0.0f;

    // Bias b1 broadcast into the WMMA accumulator (free bias add).
    v8f c0, c1;
#pragma unroll
    for (int r = 0; r < 8; ++r) { c0[r] = b1c0; c1[r] = b1c1; }

    const int nTiles  = N >> 4;                          // 16 rows per tile
    const int gwave   = blockIdx.x * (blockDim.x >> 5) + wave;
    const int wstride = gridDim.x * (blockDim.x >> 5);

    float acc = 0.0f;
    for (int t = gwave; t < nTiles; t += wstride) {
        // A matrix 16x4 f32 (2 VGPRs): lanes 0-15 row M=lane = {x0, x1},
        // lanes 16-31 hold K=2,3 = 0 padding.
        v2f a;
        if (lo) {
            const float* p = input + (((size_t)t << 4) + (size_t)j) * 2;
            a.x = p[0];
            a.y = p[1];
        } else {
            a.x = 0.0f;
            a.y = 0.0f;
        }

        v8f d0 = __builtin_amdgcn_wmma_f32_16x16x4_f32(
            false, a, false, bm0, (short)0, c0, false, false);
        v8f d1 = __builtin_amdgcn_wmma_f32_16x16x4_f32(
            false, a, false, bm1, (short)0, c1, false, false);

#pragma unroll
        for (int r = 0; r < 8; ++r) {
            // block jb0 chain
            float h1 = d0[r];                       // W1.x + b1 (from WMMA)
            float l1 = fmaxf(h1, 0.01f * h1);       // leaky_relu
            float h2 = fmaf(w2c0, l1, b2c0);
            float l2 = fmaxf(h2, 0.01f * h2);
            acc = fmaf(cc0, l2, acc);               // += w_k*W3_k * l2
            // block jb1 chain
            float g1 = d1[r];
            float m1 = fmaxf(g1, 0.01f * g1);
            float g2 = fmaf(w2c1, m1, b2c1);
            float m2 = fmaxf(g2, 0.01f * g2);
            acc = fmaf(cc1, m2, acc);
        }
    }

    // wave32 butterfly reduction
#pragma unroll
    for (int off = 16; off > 0; off >>= 1)
        acc += __shfl_xor(acc, off, 32);

    __shared__ float sred[8];
    if (lane == 0) sred[wave] = acc;
    __syncthreads();
    if (threadIdx.x == 0) {
        float s = 0.0f;
        const int nw = (int)(blockDim.x >> 5);
        for (int w = 0; w < nw; ++w) s += sred[w];
        partials[blockIdx.x] = s;
    }
}

__global__ void __launch_bounds__(256) final_reduce_kernel(
    const float* __restrict__ partials, int nPart,
    const float* __restrict__ B3, int N, int K,
    float* __restrict__ out)
{
    __shared__ float sm[256];
    float s = 0.0f;
    for (int i = threadIdx.x; i < nPart; i += 256) s += partials[i];
    sm[threadIdx.x] = s;
    __syncthreads();
    for (int stride = 128; stride > 0; stride >>= 1) {
        if ((int)threadIdx.x < stride) sm[threadIdx.x] += sm[threadIdx.x + stride];
        __syncthreads();
    }
    if (threadIdx.x == 0) {
        // constant term: N * sum_k 2*(K-k)*b3_k
        float bias = 0.0f;
        for (int k = 0; k < K; ++k) bias += 2.0f * (float)(K - k) * B3[k];
        out[0] = sm[0] + bias * (float)N;
    }
}

extern "C" void kernel_launch(void* const* d_in, const int* in_sizes, int n_in,
                              void* d_out, int out_size, void* d_ws, size_t ws_size,
                              hipStream_t stream) {
    // inputs: 0..3 unused scalars, 4=input_(N,2), 5=W1(K,1,2), 6=b1(K,1),
    //         7=W2(K,1,1), 8=b2(K,1), 9=W3(K,1,1), 10=b3(K,1)
    const float* input = (const float*)d_in[4];
    const float* W1    = (const float*)d_in[5];
    const float* B1    = (const float*)d_in[6];
    const float* W2    = (const float*)d_in[7];
    const float* B2    = (const float*)d_in[8];
    const float* W3    = (const float*)d_in[9];
    const float* B3    = (const float*)d_in[10];

    const int N = in_sizes[4] / 2;
    const int K = in_sizes[6];          // 30; kernel supports K <= 32

    int nblk = 1024;
    size_t need = (size_t)nblk * sizeof(float);
    if (ws_size < need) {
        nblk = (int)(ws_size / sizeof(float));
        if (nblk < 1) nblk = 1;
    }
    float* partials = (float*)d_ws;

    mlp_wmma_kernel<<<nblk, 256, 0, stream>>>(input, W1, B1, W2, B2, W3,
                                              partials, N, K);
    final_reduce_kernel<<<1, 256, 0, stream>>>(partials, nblk, B3, N, K,
                                               (float*)d_out);
}